// FullModelECG_60327110640159
// MI455X (gfx1250) — compile-verified
//
#include <hip/hip_runtime.h>
#include <cmath>

typedef __attribute__((ext_vector_type(2))) float v2f;
typedef __attribute__((ext_vector_type(8))) float v8f;

#define NBATCH 4
#define TILE_P 1024

__device__ __forceinline__ float waveReduceSum(float v) {
  #pragma unroll
  for (int off = 16; off > 0; off >>= 1) v += __shfl_xor(v, off, 32);
  return v;
}

// ---------------------------------------------------------------- zero init
__global__ void zero_kernel(float* __restrict__ p, int n) {
  int t = blockIdx.x * blockDim.x + threadIdx.x;
  if (t < n) p[t] = 0.0f;
}

// ---------------------------------------------------------------- FPS
// One workgroup per batch. Matches jnp scan: records current 'far' each step,
// argmax tie-break = lowest index (jnp.argmax takes first occurrence).
__global__ void fps_kernel(const float* __restrict__ pts, int N, int npoint,
                           int* __restrict__ idx) {
  __shared__ float sd[4096];
  __shared__ float rv[256];
  __shared__ int   ri[256];
  const int b = blockIdx.x;
  const int tid = threadIdx.x;
  const float* p = pts + (size_t)b * N * 3;
  for (int i = tid; i < N; i += 256) sd[i] = 1e10f;
  __syncthreads();
  int far = 0;
  for (int it = 0; it < npoint; ++it) {
    if (tid == 0) idx[b * npoint + it] = far;
    float cx = p[far * 3 + 0], cy = p[far * 3 + 1], cz = p[far * 3 + 2];
    float bm = -1.0f;
    int bi = 0x7fffffff;
    for (int i = tid; i < N; i += 256) {
      float dx = p[i * 3 + 0] - cx, dy = p[i * 3 + 1] - cy, dz = p[i * 3 + 2] - cz;
      float d = dx * dx + dy * dy + dz * dz;
      float nd = fminf(sd[i], d);
      sd[i] = nd;
      if (nd > bm) { bm = nd; bi = i; }   // ascending i keeps first max
    }
    rv[tid] = bm; ri[tid] = bi;
    __syncthreads();
    for (int s = 128; s > 0; s >>= 1) {
      if (tid < s) {
        float v2 = rv[tid + s]; int i2 = ri[tid + s];
        if (v2 > rv[tid] || (v2 == rv[tid] && i2 < ri[tid])) { rv[tid] = v2; ri[tid] = i2; }
      }
      __syncthreads();
    }
    far = ri[0];
    __syncthreads();
  }
}

// ---------------------------------------------------------------- gather
__global__ void gather_kernel(const float* __restrict__ pts, const int* __restrict__ idx,
                              int N, int npoint, float* __restrict__ out) {
  int t = blockIdx.x * blockDim.x + threadIdx.x;
  int total = NBATCH * npoint;
  if (t >= total) return;
  int b = t / npoint;
  int src = idx[t];
  const float* s = pts + ((size_t)b * N + src) * 3;
  float* d = out + (size_t)t * 3;
  d[0] = s[0]; d[1] = s[1]; d[2] = s[2];
}

// ---------------------------------------------------------------- Sinkhorn half step (WMMA)
// out[col of Q] = eps*loga - eps * lse_over_P( (pot[P] - C)/eps ),
// C = max(|p|^2 + |q|^2 - 2 p.q, 0) built by V_WMMA_F32_16X16X4_F32:
//   A row M = (p.x p.y p.z 1), B col N = (-2q.x -2q.y -2q.z |q|^2), acc = |p|^2.
// Wave owns 16 columns; rows streamed through a 1024-point LDS tile.
__global__ void sinkhorn_half_kernel(const float* __restrict__ Ppts,
                                     const float* __restrict__ Qpts,
                                     const float* __restrict__ pot,
                                     float* __restrict__ out,
                                     int N, float inv_eps, float neg_eps, float eps_loga) {
  __shared__ float4 sP[TILE_P];
  __shared__ float  sF[TILE_P];
  const int wgPerBatch = N / 128;
  const int b       = blockIdx.x / wgPerBatch;
  const int colbase = (blockIdx.x % wgPerBatch) * 128 + (threadIdx.x >> 5) * 16;
  const int lane = threadIdx.x & 31;
  const int ln15 = lane & 15;
  const bool hi  = lane >= 16;
  const int off  = hi ? 8 : 0;
  const int col  = colbase + ln15;
  const float* Pb = Ppts + (size_t)b * N * 3;
  const float* Qb = Qpts + (size_t)b * N * 3;
  const float* Fb = pot + (size_t)b * N;

  float qx = Qb[col * 3 + 0], qy = Qb[col * 3 + 1], qz = Qb[col * 3 + 2];
  float qs = qx * qx + qy * qy + qz * qz;
  v2f bfrag;
  bfrag[0] = hi ? (-2.0f * qz) : (-2.0f * qx);   // K=2 | K=0
  bfrag[1] = hi ? qs : (-2.0f * qy);             // K=3 | K=1

  float m = -INFINITY, ssum = 0.0f;

  for (int tb = 0; tb < N; tb += TILE_P) {
    __syncthreads();
    for (int i = threadIdx.x; i < TILE_P; i += 256) {
      int gi = tb + i;
      float x = Pb[gi * 3 + 0], y = Pb[gi * 3 + 1], z = Pb[gi * 3 + 2];
      sP[i] = make_float4(x, y, z, x * x + y * y + z * z);
      sF[i] = Fb[gi];
    }
    __syncthreads();
    for (int c = 0; c < TILE_P; c += 16) {
      float4 pa = sP[c + ln15];
      v2f a;
      a[0] = hi ? pa.z : pa.x;     // K=2 | K=0
      a[1] = hi ? 1.0f : pa.y;     // K=3 | K=1
      v8f acc;
      #pragma unroll
      for (int k = 0; k < 8; ++k) acc[k] = sP[c + off + k].w;
      v8f d = __builtin_amdgcn_wmma_f32_16x16x4_f32(
          false, a, false, bfrag, (short)0, acc, false, false);
      float vals[8];
      float cm = -INFINITY;
      #pragma unroll
      for (int k = 0; k < 8; ++k) {
        float C = fmaxf(d[k], 0.0f);
        float v = (sF[c + off + k] - C) * inv_eps;
        vals[k] = v;
        cm = fmaxf(cm, v);
      }
      float nm = fmaxf(m, cm);
      ssum *= __expf(m - nm);
      #pragma unroll
      for (int k = 0; k < 8; ++k) ssum += __expf(vals[k] - nm);
      m = nm;
    }
  }
  // lanes L and L+16 hold the two 8-row halves of the same column
  float m2 = __shfl_xor(m, 16, 32);
  float s2 = __shfl_xor(ssum, 16, 32);
  float M = fmaxf(m, m2);
  float S = ssum * __expf(m - M) + s2 * __expf(m2 - M);
  if (!hi) out[(size_t)b * N + col] = eps_loga + neg_eps * (M + __logf(S));
}

// ---------------------------------------------------------------- EMD final (WMMA)
// Columns = X points (potential f), rows = Y points (potential g).
// dist_i = N * sum_j exp((f_i + g_j - C_ij)/eps) * C_ij ; emd += mean_i sqrt(dist_i)
__global__ void emd_final_kernel(const float* __restrict__ Xpts,
                                 const float* __restrict__ Ypts,
                                 const float* __restrict__ f,
                                 const float* __restrict__ g,
                                 int N, float inv_eps, float* __restrict__ emd_out) {
  __shared__ float4 sP[TILE_P];
  __shared__ float  sG[TILE_P];
  const int wgPerBatch = N / 128;
  const int b       = blockIdx.x / wgPerBatch;
  const int colbase = (blockIdx.x % wgPerBatch) * 128 + (threadIdx.x >> 5) * 16;
  const int lane = threadIdx.x & 31;
  const int ln15 = lane & 15;
  const bool hi  = lane >= 16;
  const int off  = hi ? 8 : 0;
  const int col  = colbase + ln15;
  const float* Yb = Ypts + (size_t)b * N * 3;
  const float* Xb = Xpts + (size_t)b * N * 3;
  const float* Gb = g + (size_t)b * N;

  float qx = Xb[col * 3 + 0], qy = Xb[col * 3 + 1], qz = Xb[col * 3 + 2];
  float qs = qx * qx + qy * qy + qz * qz;
  v2f bfrag;
  bfrag[0] = hi ? (-2.0f * qz) : (-2.0f * qx);
  bfrag[1] = hi ? qs : (-2.0f * qy);
  const float fcol = f[(size_t)b * N + col];

  float accsum = 0.0f;

  for (int tb = 0; tb < N; tb += TILE_P) {
    __syncthreads();
    for (int i = threadIdx.x; i < TILE_P; i += 256) {
      int gi = tb + i;
      float x = Yb[gi * 3 + 0], y = Yb[gi * 3 + 1], z = Yb[gi * 3 + 2];
      sP[i] = make_float4(x, y, z, x * x + y * y + z * z);
      sG[i] = Gb[gi];
    }
    __syncthreads();
    for (int c = 0; c < TILE_P; c += 16) {
      float4 pa = sP[c + ln15];
      v2f a;
      a[0] = hi ? pa.z : pa.x;
      a[1] = hi ? 1.0f : pa.y;
      v8f acc;
      #pragma unroll
      for (int k = 0; k < 8; ++k) acc[k] = sP[c + off + k].w;
      v8f d = __builtin_amdgcn_wmma_f32_16x16x4_f32(
          false, a, false, bfrag, (short)0, acc, false, false);
      #pragma unroll
      for (int k = 0; k < 8; ++k) {
        float C = fmaxf(d[k], 0.0f);
        accsum += __expf((sG[c + off + k] + fcol - C) * inv_eps) * C;
      }
    }
  }
  accsum += __shfl_xor(accsum, 16, 32);
  if (!hi) {
    float dist = (float)N * accsum;
    atomicAdd(&emd_out[b], sqrtf(dist) / (float)N);
  }
}

// ---------------------------------------------------------------- Chamfer one direction
// For each point of A, min sqdist over B; accumulate sqrt(d)*scale_p and d*scale_t.
__global__ void chamfer_min_kernel(const float* __restrict__ A, int NA,
                                   const float* __restrict__ Bp, int NB,
                                   float* __restrict__ out_p, float* __restrict__ out_t,
                                   float scale_p, float scale_t) {
  __shared__ float sx[256], sy[256], sz[256];
  const int wg = NA / 256;
  const int b  = blockIdx.x / wg;
  const int ai = (blockIdx.x % wg) * 256 + threadIdx.x;
  const float* Ab = A + (size_t)b * NA * 3;
  const float* Bb = Bp + (size_t)b * NB * 3;
  float ax = Ab[ai * 3 + 0], ay = Ab[ai * 3 + 1], az = Ab[ai * 3 + 2];
  float minv = 1e30f;
  for (int tb = 0; tb < NB; tb += 256) {
    __syncthreads();
    int gi = tb + threadIdx.x;
    sx[threadIdx.x] = Bb[gi * 3 + 0];
    sy[threadIdx.x] = Bb[gi * 3 + 1];
    sz[threadIdx.x] = Bb[gi * 3 + 2];
    __syncthreads();
    #pragma unroll 8
    for (int j = 0; j < 256; ++j) {
      float dx = ax - sx[j], dy = ay - sy[j], dz = az - sz[j];
      minv = fminf(minv, dx * dx + dy * dy + dz * dz);
    }
  }
  atomicAdd(&out_p[b], sqrtf(minv) * scale_p);
  atomicAdd(&out_t[b], minv * scale_t);
}

// ---------------------------------------------------------------- uniform loss
// One wave per (batch, seed, percentage-launch). First-nsample-in-index-order
// selection within radius (ballot compaction), pad with first selection,
// nearest-intra-group distance, accumulate weighted norm.
__global__ void uniform_kernel(const float* __restrict__ pcd,
                               const float* __restrict__ seeds,
                               int N, int npoint, int nsample, float rr,
                               float expect_len, float wcoef,
                               float* __restrict__ u_out) {
  __shared__ int sel[64];
  __shared__ float gx[64], gy[64], gz[64];
  const int b = blockIdx.x / npoint;
  const int s = blockIdx.x % npoint;
  const int lane = threadIdx.x;
  const float* pb = pcd + (size_t)b * N * 3;
  const float* sd = seeds + ((size_t)b * npoint + s) * 3;
  const float cx = sd[0], cy = sd[1], cz = sd[2];

  int count = 0;
  for (int base = 0; base < N && count < nsample; base += 32) {
    int i = base + lane;
    float dx = pb[i * 3 + 0] - cx, dy = pb[i * 3 + 1] - cy, dz = pb[i * 3 + 2] - cz;
    float d2 = dx * dx + dy * dy + dz * dz;
    bool in = d2 < rr;
    unsigned mask = (unsigned)__ballot(in);
    int prefix = __popc(mask & ((1u << lane) - 1u));
    int pos = count + prefix;
    if (in && pos < nsample) sel[pos] = i;
    count += __popc(mask);
  }
  __syncthreads();
  if (count > nsample) count = nsample;
  for (int t = count + lane; t < nsample; t += 32) sel[t] = sel[0];
  __syncthreads();
  for (int t = lane; t < nsample; t += 32) {
    int id = sel[t];
    gx[t] = pb[id * 3 + 0]; gy[t] = pb[id * 3 + 1]; gz[t] = pb[id * 3 + 2];
  }
  __syncthreads();
  float sumd = 0.0f;
  for (int t = lane; t < nsample; t += 32) {
    float px = gx[t], py = gy[t], pz = gz[t];
    float minv = 1e10f;                       // matches +eye*1e10 cap
    for (int j = 0; j < nsample; ++j) {
      if (j == t) continue;
      float dx = px - gx[j], dy = py - gy[j], dz = pz - gz[j];
      minv = fminf(minv, dx * dx + dy * dy + dz * dz);
    }
    sumd += sqrtf(fabsf(minv + 1e-8f));
  }
  sumd = waveReduceSum(sumd);
  if (lane == 0) {
    float mean_d = sumd / (float)nsample;
    float diff = mean_d - expect_len;
    float norm = diff * diff / (expect_len + 1e-8f);
    atomicAdd(u_out, norm * wcoef);
  }
}

// ---------------------------------------------------------------- launch
extern "C" void kernel_launch(void* const* d_in, const int* in_sizes, int n_in,
                              void* d_out, int out_size, void* d_ws, size_t ws_size,
                              hipStream_t stream) {
  (void)in_sizes; (void)n_in; (void)out_size; (void)ws_size;
  const float* out1 = (const float*)d_in[0];  // (4,1024,3)
  const float* out2 = (const float*)d_in[1];  // (4,4096,3)
  const float* gt   = (const float*)d_in[2];  // (4,4096,3)
  float* O = (float*)d_out;

  const int B = NBATCH, N1 = 1024, N2 = 4096;
  float* ws     = (float*)d_ws;
  float* gt_fps = ws;                   // 12288
  float* f1 = gt_fps + B * N1 * 3;      // 4096
  float* g1 = f1 + B * N1;              // 4096
  float* f2 = g1 + B * N1;              // 16384
  float* g2 = f2 + B * N2;              // 16384
  float* seeds1 = g2 + B * N2;          // 4*51*3  = 612
  float* seeds2 = seeds1 + 612;         // 4*204*3 = 2448
  int*   idxbuf = (int*)(seeds2 + 2448 + 4);  // up to 4096 ints

  float* emd1  = O + 61440;
  float* emd2  = O + 61444;
  float* cd_p1 = O + 61448;
  float* cd_p2 = O + 61452;
  float* cd_t1 = O + 61456;
  float* cd_t2 = O + 61460;
  float* u1    = O + 61464;
  float* u2    = O + 61465;

  // passthrough outputs
  (void)hipMemcpyAsync(O, out1, sizeof(float) * B * N1 * 3, hipMemcpyDeviceToDevice, stream);
  (void)hipMemcpyAsync(O + 12288, out2, sizeof(float) * B * N2 * 3, hipMemcpyDeviceToDevice, stream);

  // zero accumulators and Sinkhorn potentials
  zero_kernel<<<1, 32, 0, stream>>>(O + 61440, 26);
  {
    int npot = B * N1 * 2 + B * N2 * 2;  // 40960
    zero_kernel<<<(npot + 255) / 256, 256, 0, stream>>>(f1, npot);
  }

  // FPS(gt, 1024) -> gt_fps
  fps_kernel<<<B, 256, 0, stream>>>(gt, N2, N1, idxbuf);
  gather_kernel<<<(B * N1 + 255) / 256, 256, 0, stream>>>(gt, idxbuf, N2, N1, gt_fps);

  const float eps = 0.005f;
  const float inv_eps = 1.0f / eps;
  const float eps_loga1 = -eps * logf((float)N1);
  const float eps_loga2 = -eps * logf((float)N2);

  // EMD 1: x=output1 vs y=gt_fps
  for (int it = 0; it < 10; ++it) {
    sinkhorn_half_kernel<<<B * (N1 / 128), 256, 0, stream>>>(out1, gt_fps, f1, g1, N1, inv_eps, -eps, eps_loga1);
    sinkhorn_half_kernel<<<B * (N1 / 128), 256, 0, stream>>>(gt_fps, out1, g1, f1, N1, inv_eps, -eps, eps_loga1);
  }
  emd_final_kernel<<<B * (N1 / 128), 256, 0, stream>>>(out1, gt_fps, f1, g1, N1, inv_eps, emd1);

  // EMD 2: x=output2 vs y=gt
  for (int it = 0; it < 10; ++it) {
    sinkhorn_half_kernel<<<B * (N2 / 128), 256, 0, stream>>>(out2, gt, f2, g2, N2, inv_eps, -eps, eps_loga2);
    sinkhorn_half_kernel<<<B * (N2 / 128), 256, 0, stream>>>(gt, out2, g2, f2, N2, inv_eps, -eps, eps_loga2);
  }
  emd_final_kernel<<<B * (N2 / 128), 256, 0, stream>>>(out2, gt, f2, g2, N2, inv_eps, emd2);

  // Chamfer: d11 (per-gt min over out1), d12 (per-out1 min over gt)
  chamfer_min_kernel<<<B * (N2 / 256), 256, 0, stream>>>(gt, N2, out1, N1, cd_p1, cd_t1, 0.5f / N2, 1.0f / N2);
  chamfer_min_kernel<<<B * (N1 / 256), 256, 0, stream>>>(out1, N1, gt, N2, cd_p1, cd_t1, 0.5f / N1, 1.0f / N1);
  chamfer_min_kernel<<<B * (N2 / 256), 256, 0, stream>>>(gt, N2, out2, N2, cd_p2, cd_t2, 0.5f / N2, 1.0f / N2);
  chamfer_min_kernel<<<B * (N2 / 256), 256, 0, stream>>>(out2, N2, gt, N2, cd_p2, cd_t2, 0.5f / N2, 1.0f / N2);

  // uniform loss seeds
  const int np1 = N1 / 20;  // 51
  const int np2 = N2 / 20;  // 204
  fps_kernel<<<B, 256, 0, stream>>>(out1, N1, np1, idxbuf);
  gather_kernel<<<(B * np1 + 255) / 256, 256, 0, stream>>>(out1, idxbuf, N1, np1, seeds1);
  fps_kernel<<<B, 256, 0, stream>>>(out2, N2, np2, idxbuf);
  gather_kernel<<<(B * np2 + 255) / 256, 256, 0, stream>>>(out2, idxbuf, N2, np2, seeds2);

  const double PCT[5] = {0.004, 0.006, 0.008, 0.01, 0.012};
  const double PI_D = 3.14159265358979323846;
  for (int pi = 0; pi < 5; ++pi) {
    double p = PCT[pi];
    {
      int ns = (int)((double)N1 * p);
      float el = (float)sqrt(PI_D * p / (double)ns);
      float wc = (float)((p * 100.0) * (p * 100.0) / (double)(B * np1 * 5));
      uniform_kernel<<<B * np1, 32, 0, stream>>>(out1, seeds1, N1, np1, ns, (float)p, el, wc, u1);
    }
    {
      int ns = (int)((double)N2 * p);
      float el = (float)sqrt(PI_D * p / (double)ns);
      float wc = (float)((p * 100.0) * (p * 100.0) / (double)(B * np2 * 5));
      uniform_kernel<<<B * np2, 32, 0, stream>>>(out2, seeds2, N2, np2, ns, (float)p, el, wc, u2);
    }
  }
}